// SubGraphAvgPool_80367428043175
// MI455X (gfx1250) — compile-verified
//
#include <hip/hip_runtime.h>
#include <cstdint>

// SubGraphAvgPool, B=16, N=8193, D=512, m=4 -> G=2048.
// out[b,g,d] = 0.2*(h[b,g,d] + h[b,4g+1,d] + h[b,4g+2,d] + h[b,4g+3,d] + h[b,4g+4,d])
//
// Pure bandwidth op (~384 MB moved, ~84 MFLOP). Strategy: per-wave double-
// buffered async global->LDS pipeline (CDNA5 ASYNCcnt path), b128 everywhere.

#define DFEAT   512
#define NNODES  8193
#define GSUB    2048
#define BATCH   16
#define NG      8                       // g's per wave
#define WAVES_PB 2                      // waves per block (64 threads)
#define CHUNK_FLOATS (5 * DFEAT)        // 2560 floats = 10240 B per g
#define CHUNK_BYTES  (CHUNK_FLOATS * 4)
#define ISSUES_HEAD 4                   // 2048 B / (32 lanes * 16 B)
#define ISSUES_BODY 16                  // 8192 B / 512 B
#define ISSUES_PER_CHUNK (ISSUES_HEAD + ISSUES_BODY)  // 20

// Issue N async 512B wave-copies (b128 per lane) of a contiguous region.
// The instruction's 24-bit offset is added to BOTH the global and LDS address
// (CDNA5 ISA load pseudocode), so one address pair suffices.
template <int T, int N>
struct AsyncCp {
  static __device__ __forceinline__ void run(uint32_t la, uint64_t ga) {
    asm volatile("global_load_async_to_lds_b128 %0, %1, off offset:%c2"
                 :: "v"(la), "v"(ga), "n"(T * 512) : "memory");
    AsyncCp<T + 1, N>::run(la, ga);
  }
};
template <int N>
struct AsyncCp<N, N> {
  static __device__ __forceinline__ void run(uint32_t, uint64_t) {}
};

__global__ __launch_bounds__(WAVES_PB * 32)
void subgraph_avgpool_kernel(const float* __restrict__ h, float* __restrict__ out) {
  __shared__ __align__(16) float smem[WAVES_PB * 2 * CHUNK_FLOATS];  // 40 KB

  const int lane = (int)(threadIdx.x & 31u);
  const int wv   = (int)(threadIdx.x >> 5);
  const int b    = (int)blockIdx.y;
  const int g0   = (int)blockIdx.x * (WAVES_PB * NG) + wv * NG;

  const float* hb = h   + (size_t)b * NNODES * DFEAT;
  float*       ob = out + (size_t)b * GSUB   * DFEAT;

  // Workgroup-relative LDS byte offset of this wave's buffer pair
  // (low 32 bits of the generic pointer to __shared__ data).
  float* wbuf = &smem[wv * 2 * CHUNK_FLOATS];
  const uint32_t lbase = (uint32_t)(uintptr_t)(void*)wbuf;

  // Issue one 10KB chunk (head row 2KB + body rows 8KB) into buffer (i&1).
  auto issue = [&](int i) {
    const int g = g0 + i;
    const uint32_t lb = lbase + (uint32_t)((i & 1) * CHUNK_BYTES)
                              + (uint32_t)(lane * 16);
    const float* head = hb + (size_t)g * DFEAT;
    const float* body = hb + (size_t)(4 * g + 1) * DFEAT;
    const uint64_t gah = (uint64_t)(uintptr_t)head + (uint32_t)(lane * 16);
    const uint64_t gab = (uint64_t)(uintptr_t)body + (uint32_t)(lane * 16);
    AsyncCp<0, ISSUES_HEAD>::run(lb, gah);          // LDS [0    .. 2048)
    AsyncCp<0, ISSUES_BODY>::run(lb + 2048, gab);   // LDS [2048 .. 10240)
  };

  // Prologue: two chunks in flight (ASYNCcnt = 40, counter max 63).
  issue(0);
  issue(1);

#pragma unroll
  for (int i = 0; i < NG; ++i) {
    // Async loads retire in order: cnt<=20 means chunk i is fully in LDS
    // (only chunk i+1's 20 issues may remain). Last iteration drains fully.
    if (i == NG - 1) {
      asm volatile("s_wait_asynccnt 0" ::: "memory");
    } else {
      asm volatile("s_wait_asynccnt %c0" :: "n"(ISSUES_PER_CHUNK) : "memory");
    }

    const float* buf = wbuf + (i & 1) * CHUNK_FLOATS;
    float* orow = ob + (size_t)(g0 + i) * DFEAT;

#pragma unroll
    for (int j = 0; j < 4; ++j) {
      const int off = j * 128 + lane * 4;   // 16B-aligned float4 slots
      float4 hd = *(const float4*)(buf + off);
      float4 b0 = *(const float4*)(buf +  512 + off);
      float4 b1 = *(const float4*)(buf + 1024 + off);
      float4 b2 = *(const float4*)(buf + 1536 + off);
      float4 b3 = *(const float4*)(buf + 2048 + off);
      float4 r;
      r.x = 0.2f * (hd.x + b0.x + b1.x + b2.x + b3.x);
      r.y = 0.2f * (hd.y + b0.y + b1.y + b2.y + b3.y);
      r.z = 0.2f * (hd.z + b0.z + b1.z + b2.z + b3.z);
      r.w = 0.2f * (hd.w + b0.w + b1.w + b2.w + b3.w);
      *(float4*)(orow + off) = r;
    }

    // Refill the buffer we just consumed (ds reads already waited on by the
    // VALU consumers above, so the WAR on LDS is resolved before overwrite).
    if (i + 2 < NG) issue(i + 2);
  }
}

extern "C" void kernel_launch(void* const* d_in, const int* in_sizes, int n_in,
                              void* d_out, int out_size, void* d_ws, size_t ws_size,
                              hipStream_t stream) {
  (void)in_sizes; (void)n_in; (void)out_size; (void)d_ws; (void)ws_size;
  const float* h  = (const float*)d_in[0];   // [16, 8193, 512] f32
  float*       out = (float*)d_out;          // [16, 2048, 512] f32
  // d_in[1] is m (=4); the index structure for m=4 is hardcoded above.
  dim3 grid(GSUB / (WAVES_PB * NG), BATCH, 1);   // 128 x 16
  dim3 block(WAVES_PB * 32, 1, 1);               // 64 threads = 2 waves
  subgraph_avgpool_kernel<<<grid, block, 0, stream>>>(h, out);
}